// MoE_Compressor_69501160784684
// MI455X (gfx1250) — compile-verified
//
#include <hip/hip_runtime.h>
#include <math.h>

typedef float v2f __attribute__((ext_vector_type(2)));
typedef float v8f __attribute__((ext_vector_type(8)));

#define D_IN   1024
#define NSLOT  64
#define HID    512
#define NCHUNK 48

// ---------------------------------------------------------------------------
// Kernel 1: gate logits via V_WMMA_F32_16X16X4_F32, softmax denom + top-2.
// One wave handles 16 tokens x 64 slots (4 WMMA accumulators), K loop = 1024.
// ---------------------------------------------------------------------------
__global__ __launch_bounds__(256) void gate_topk_kernel(
    const float* __restrict__ x, const float* __restrict__ gw,
    const float* __restrict__ gb, int N,
    int2* __restrict__ idxs, float2* __restrict__ wts)
{
    __shared__ float lds[8][16][68];   // [wave][token][slot] padded vs bank conflicts

    const int tid  = threadIdx.x;
    const int wave = tid >> 5;
    const int lane = tid & 31;
    const int half = lane >> 4;        // 0: K=0,1   1: K=2,3   (A/B layout)
    const int l16  = lane & 15;
    const int tile_base = blockIdx.x * 128 + wave * 16;

    int row = tile_base + l16;
    if (row >= N) row = N - 1;                 // clamp: keep EXEC all-ones for WMMA
    const float* xr = x + (size_t)row * D_IN;
    const int kp = half * 2;

    v8f c0 = {}, c1 = {}, c2 = {}, c3 = {};
    #pragma unroll 4
    for (int k0 = 0; k0 < D_IN; k0 += 4) {
        // A fragment: lane(M=l16) holds K=kp, kp+1 (contiguous 8B load)
        v2f a = *(const v2f*)(xr + k0 + kp);
        // B fragments: lane(N=l16) holds gate_w[K=kp][n], gate_w[K=kp+1][n]
        const float* g0 = gw + (size_t)(k0 + kp) * NSLOT + l16;
        const float* g1 = g0 + NSLOT;
        v2f b0 = { g0[0],  g1[0]  };
        v2f b1f = { g0[16], g1[16] };
        v2f b2f = { g0[32], g1[32] };
        v2f b3f = { g0[48], g1[48] };
        c0 = __builtin_amdgcn_wmma_f32_16x16x4_f32(false, a, false, b0,  (short)0, c0, false, false);
        c1 = __builtin_amdgcn_wmma_f32_16x16x4_f32(false, a, false, b1f, (short)0, c1, false, false);
        c2 = __builtin_amdgcn_wmma_f32_16x16x4_f32(false, a, false, b2f, (short)0, c2, false, false);
        c3 = __builtin_amdgcn_wmma_f32_16x16x4_f32(false, a, false, b3f, (short)0, c3, false, false);
    }

    // C layout: VGPR j -> token M = j + 8*half, slot N = (frag*16 + l16)
    #pragma unroll
    for (int j = 0; j < 8; ++j) {
        int tok = j + 8 * half;
        lds[wave][tok][ 0 + l16] = c0[j];
        lds[wave][tok][16 + l16] = c1[j];
        lds[wave][tok][32 + l16] = c2[j];
        lds[wave][tok][48 + l16] = c3[j];
    }
    __syncthreads();

    if (lane < 16) {
        int token = tile_base + lane;
        if (token < N) {
            // pass 1: max logit
            float mx = -3.4e38f;
            for (int s = 0; s < NSLOT; ++s) {
                float l = lds[wave][lane][s] + gb[s];
                mx = fmaxf(mx, l);
            }
            // pass 2: softmax denom + top-2 (strict > keeps earliest index on ties)
            float denom = 0.f;
            float m1 = -3.4e38f, m2 = -3.4e38f;
            int   i1 = 0, i2 = 0;
            for (int s = 0; s < NSLOT; ++s) {
                float l = lds[wave][lane][s] + gb[s];
                denom += __expf(l - mx);
                if (l > m1)      { m2 = m1; i2 = i1; m1 = l; i1 = s; }
                else if (l > m2) { m2 = l; i2 = s; }
            }
            float inv = 1.f / denom;
            wts[token]  = make_float2(__expf(m1 - mx) * inv, __expf(m2 - mx) * inv);
            idxs[token] = make_int2(i1, i2);
        }
    }
}

// ---------------------------------------------------------------------------
// Kernel 2: weighted aggregation. grid = (NCHUNK chunks, 8 slot-groups of 8).
// Accumulate into 32KB LDS, write a private partial (no atomics -> deterministic).
// ---------------------------------------------------------------------------
__global__ __launch_bounds__(256) void aggregate_kernel(
    const float* __restrict__ x, const int2* __restrict__ idxs,
    const float2* __restrict__ wts, int N,
    float* __restrict__ partial, float* __restrict__ pcnt)
{
    __shared__ float acc[8 * 1024];
    __shared__ float cnt[8];
    const int tid   = threadIdx.x;
    const int chunk = blockIdx.x;
    const int group = blockIdx.y;

    const int per = (N + NCHUNK - 1) / NCHUNK;
    const int t0 = chunk * per;
    const int t1 = min(N, t0 + per);

    for (int i = tid; i < 8 * 1024; i += 256) acc[i] = 0.f;
    if (tid < 8) cnt[tid] = 0.f;
    __syncthreads();

    const int d0 = tid * 4;
    for (int t = t0; t < t1; ++t) {
        int2   id = idxs[t];          // uniform across block -> scalar broadcast
        float2 w  = wts[t];
        bool m0 = (id.x >> 3) == group;
        bool m1 = (id.y >> 3) == group;
        if (m0 || m1) {
            float4 v = *(const float4*)(x + (size_t)t * D_IN + d0);
            if (m0) {
                int s = (id.x & 7) * 1024 + d0;
                acc[s+0] += w.x * v.x; acc[s+1] += w.x * v.y;
                acc[s+2] += w.x * v.z; acc[s+3] += w.x * v.w;
            }
            if (m1) {
                int s = (id.y & 7) * 1024 + d0;
                acc[s+0] += w.y * v.x; acc[s+1] += w.y * v.y;
                acc[s+2] += w.y * v.z; acc[s+3] += w.y * v.w;
            }
            if (tid == 0) {
                if (m0) cnt[id.x & 7] += w.x;
                if (m1) cnt[id.y & 7] += w.y;
            }
        }
    }
    __syncthreads();

    float* dst = partial + (size_t)chunk * (NSLOT * D_IN) + group * (8 * 1024);
    for (int i = tid; i < 8 * 1024; i += 256) dst[i] = acc[i];
    if (tid < 8) pcnt[chunk * NSLOT + group * 8 + tid] = cnt[tid];
}

// ---------------------------------------------------------------------------
// Kernel 3: reduce partials -> ct = compressed/(count+1e-9); write aux_loss=0.
// ---------------------------------------------------------------------------
__global__ __launch_bounds__(256) void finalize_ct_kernel(
    const float* __restrict__ partial, const float* __restrict__ pcnt,
    float* __restrict__ ct, float* __restrict__ out, int out_size)
{
    int i = blockIdx.x * 256 + threadIdx.x;      // 65536 elements
    int k = i >> 10;
    float cs = 0.f, comp = 0.f;
    #pragma unroll 4
    for (int c = 0; c < NCHUNK; ++c) {
        cs   += pcnt[c * NSLOT + k];
        comp += partial[(size_t)c * (NSLOT * D_IN) + i];
    }
    ct[i] = comp / (cs + 1e-9f);
    if (i == 0 && out_size > NSLOT * D_IN) out[NSLOT * D_IN] = 0.f;  // aux_loss
}

// ---------------------------------------------------------------------------
// Kernel 4a: h = GELU(ct_k @ w1_k + b1_k)   (exact erf GELU)
// grid = (64 slots, 2), block = 256 -> one h element per thread.
// ---------------------------------------------------------------------------
__global__ __launch_bounds__(256) void expert_h_kernel(
    const float* __restrict__ ct, const float* __restrict__ w1,
    const float* __restrict__ b1, float* __restrict__ hbuf)
{
    __shared__ float cts[D_IN];
    const int k = blockIdx.x;
    const int h = blockIdx.y * 256 + threadIdx.x;
    for (int i = threadIdx.x; i < D_IN; i += 256) cts[i] = ct[k * D_IN + i];
    __syncthreads();

    const float* w = w1 + (size_t)k * D_IN * HID + h;
    float a = b1[k * HID + h];
    #pragma unroll 8
    for (int d = 0; d < D_IN; ++d) a += cts[d] * w[(size_t)d * HID];
    hbuf[k * HID + h] = 0.5f * a * (1.f + erff(a * 0.70710678118654752f));
}

// ---------------------------------------------------------------------------
// Kernel 4b: y = h_k @ w2_k + b2_k; LayerNorm (population var); residual.
// grid = 64 slots, block = 256, 4 outputs/thread.
// ---------------------------------------------------------------------------
__global__ __launch_bounds__(256) void expert_out_kernel(
    const float* __restrict__ ct, const float* __restrict__ hbuf,
    const float* __restrict__ w2, const float* __restrict__ b2,
    const float* __restrict__ lng, const float* __restrict__ lnb,
    float* __restrict__ out)
{
    __shared__ float hs[HID];
    __shared__ float redS[8], redQ[8];
    __shared__ float s_mu, s_inv;
    const int k = blockIdx.x, tid = threadIdx.x;

    for (int i = tid; i < HID; i += 256) hs[i] = hbuf[k * HID + i];
    __syncthreads();

    float y[4];
    #pragma unroll
    for (int j = 0; j < 4; ++j) y[j] = b2[k * D_IN + tid + j * 256];

    const float* w = w2 + (size_t)k * HID * D_IN;
    #pragma unroll 4
    for (int h = 0; h < HID; ++h) {
        float hv = hs[h];
        const float* wr = w + (size_t)h * D_IN + tid;
        #pragma unroll
        for (int j = 0; j < 4; ++j) y[j] += hv * wr[j * 256];
    }

    float s = y[0] + y[1] + y[2] + y[3];
    float q = y[0]*y[0] + y[1]*y[1] + y[2]*y[2] + y[3]*y[3];
    #pragma unroll
    for (int off = 16; off > 0; off >>= 1) {       // wave32 shuffle reduce
        s += __shfl_down(s, off, 32);
        q += __shfl_down(q, off, 32);
    }
    if ((tid & 31) == 0) { redS[tid >> 5] = s; redQ[tid >> 5] = q; }
    __syncthreads();
    if (tid == 0) {
        float S = 0.f, Q = 0.f;
        for (int i = 0; i < 8; ++i) { S += redS[i]; Q += redQ[i]; }
        float mu  = S * (1.f / 1024.f);
        float var = Q * (1.f / 1024.f) - mu * mu;
        s_mu = mu; s_inv = rsqrtf(var + 1e-5f);
    }
    __syncthreads();
    const float mu = s_mu, inv = s_inv;
    #pragma unroll
    for (int j = 0; j < 4; ++j) {
        int idx = k * D_IN + tid + j * 256;
        out[idx] = ct[idx] + ((y[j] - mu) * inv * lng[idx] + lnb[idx]);
    }
}

// ---------------------------------------------------------------------------
extern "C" void kernel_launch(void* const* d_in, const int* in_sizes, int n_in,
                              void* d_out, int out_size, void* d_ws, size_t ws_size,
                              hipStream_t stream) {
    const float* x      = (const float*)d_in[0];
    const float* gate_w = (const float*)d_in[1];
    const float* gate_b = (const float*)d_in[2];
    const float* w1     = (const float*)d_in[3];
    const float* b1     = (const float*)d_in[4];
    const float* w2     = (const float*)d_in[5];
    const float* b2     = (const float*)d_in[6];
    const float* ln_g   = (const float*)d_in[7];
    const float* ln_b   = (const float*)d_in[8];
    float* out = (float*)d_out;

    const int N = in_sizes[0] / D_IN;

    char* ws = (char*)d_ws;
    size_t off = 0;
    auto carve = [&](size_t bytes) { void* p = ws + off; off = (off + bytes + 255) & ~(size_t)255; return p; };
    int2*   idxs    = (int2*)  carve((size_t)N * sizeof(int2));
    float2* wtsbuf  = (float2*)carve((size_t)N * sizeof(float2));
    float*  partial = (float*) carve((size_t)NCHUNK * NSLOT * D_IN * sizeof(float));
    float*  pcnt    = (float*) carve((size_t)NCHUNK * NSLOT * sizeof(float));
    float*  ct      = (float*) carve((size_t)NSLOT * D_IN * sizeof(float));
    float*  hbuf    = (float*) carve((size_t)NSLOT * HID * sizeof(float));

    int blocks1 = (N + 127) / 128;
    gate_topk_kernel<<<blocks1, 256, 0, stream>>>(x, gate_w, gate_b, N, idxs, wtsbuf);

    dim3 grid2(NCHUNK, 8);
    aggregate_kernel<<<grid2, 256, 0, stream>>>(x, idxs, wtsbuf, N, partial, pcnt);

    finalize_ct_kernel<<<(NSLOT * D_IN) / 256, 256, 0, stream>>>(partial, pcnt, ct, out, out_size);

    dim3 grid4(NSLOT, HID / 256);
    expert_h_kernel<<<grid4, 256, 0, stream>>>(ct, w1, b1, hbuf);

    expert_out_kernel<<<NSLOT, 256, 0, stream>>>(ct, hbuf, w2, b2, ln_g, ln_b, out);
}